// GATLayer_19739669692736
// MI455X (gfx1250) — compile-verified
//
#include <hip/hip_runtime.h>
#include <hip/hip_bf16.h>

// GAT layer on MI455X (gfx1250, wave32).
// Stages 1/2/5 are fp32 GEMMs -> V_WMMA_F32_16X16X4_F32 (full f32 precision).
// Stage 3 (relu inside the contraction) is VALU with LDS tiling.
// Softmax denominator via deterministic two-level tree reduction (no atomics).

#define B_  2
#define N_  512
#define F_  768
#define ATT_ 64
#define HID_ 128

typedef __attribute__((ext_vector_type(2))) float v2f;
typedef __attribute__((ext_vector_type(8))) float v8f;

// ---------------------------------------------------------------------------
// Kernel 1: h[m, a] = sum_f x[m, f] * W_w[a, f] + W_b[a]
//   M = B*N = 1024, K = 768, Ncol = 64.  16x16 tile per wave, f32 WMMA, K step 4.
// ---------------------------------------------------------------------------
__global__ void k_h(const float* __restrict__ x, const float* __restrict__ Ww,
                    const float* __restrict__ Wb, float* __restrict__ h) {
    const int lane = threadIdx.x & 31;
    const int wave = threadIdx.x >> 5;
    const int tile = blockIdx.x * 4 + wave;   // 256 tiles = 64 M-tiles x 4 N-tiles
    const int tm = tile >> 2;
    const int tn = tile & 3;
    const int kb = (lane >> 4) * 2;           // ISA 16x4 A layout: lanes 16-31 -> K=2,3
    const int m  = tm * 16 + (lane & 15);
    const int n  = tn * 16 + (lane & 15);

    const float* xrow = x  + (size_t)m * F_;
    const float* wrow = Ww + (size_t)n * F_;  // B[k][n] = W_w[n][k]

    v8f c = {};
    for (int k = 0; k < F_; k += 4) {
        v2f a, b;
        a.x = xrow[k + kb]; a.y = xrow[k + kb + 1];
        b.x = wrow[k + kb]; b.y = wrow[k + kb + 1];
        c = __builtin_amdgcn_wmma_f32_16x16x4_f32(false, a, false, b,
                                                  (short)0, c, false, false);
    }
    const float bias = Wb[n];
    const int row0 = tm * 16 + (lane >> 4) * 8;   // D layout: VGPR v -> row v (+8 hi lanes)
    const int col  = tn * 16 + (lane & 15);
#pragma unroll
    for (int v = 0; v < 8; ++v)
        h[(size_t)(row0 + v) * ATT_ + col] = c[v] + bias;
}

// ---------------------------------------------------------------------------
// Kernel 2: e12[m, c] :  c<128 -> e1[m,c] = sum_a h[m,a]*W1[c, a]
//                        c>=128 -> e2[m,c-128] = sum_a h[m,a]*W1[c-128, 64+a]
//   M = 1024, K = 64, Ncol = 256.
// ---------------------------------------------------------------------------
__global__ void k_e12(const float* __restrict__ h, const float* __restrict__ W1,
                      float* __restrict__ e12) {
    const int lane = threadIdx.x & 31;
    const int wave = threadIdx.x >> 5;
    const int tile = blockIdx.x * 4 + wave;   // 1024 tiles = 64 x 16
    const int tm = tile >> 4;
    const int tn = tile & 15;
    const int kb = (lane >> 4) * 2;
    const int m    = tm * 16 + (lane & 15);
    const int ncol = tn * 16 + (lane & 15);                 // 0..255
    const int w1row = (ncol < HID_) ? ncol : (ncol - HID_);
    const int koff  = (ncol < HID_) ? 0 : ATT_;

    const float* hrow = h  + (size_t)m * ATT_;
    const float* wrow = W1 + (size_t)w1row * (2 * ATT_) + koff;

    v8f c = {};
    for (int k = 0; k < ATT_; k += 4) {
        v2f a, b;
        a.x = hrow[k + kb]; a.y = hrow[k + kb + 1];
        b.x = wrow[k + kb]; b.y = wrow[k + kb + 1];
        c = __builtin_amdgcn_wmma_f32_16x16x4_f32(false, a, false, b,
                                                  (short)0, c, false, false);
    }
    const int row0 = tm * 16 + (lane >> 4) * 8;
    const int col  = tn * 16 + (lane & 15);
#pragma unroll
    for (int v = 0; v < 8; ++v)
        e12[(size_t)(row0 + v) * 256 + col] = c[v];
}

// ---------------------------------------------------------------------------
// Kernel 3: expE[b,i,j] = adj ? exp(leaky(sum_h relu(e1+e2+b1)*W2 + b2)) : 0
//   32x32 (i,j) tile per 256-thread block; LDS rows padded to 129 (bank spread).
//   Each thread: 1 i row x 4 j columns (register blocking).
//   Deterministic block partial sum of exp for the softmax denominator.
// ---------------------------------------------------------------------------
__global__ void k_pair(const float* __restrict__ e12, const float* __restrict__ b1,
                       const float* __restrict__ W2, const float* __restrict__ b2,
                       const int* __restrict__ adj, float* __restrict__ expE,
                       float* __restrict__ partials) {
    __shared__ float se1[32][HID_ + 1];
    __shared__ float se2[32][HID_ + 1];
    __shared__ float sb1[HID_];
    __shared__ float sw2[HID_];
    __shared__ float red[256];

    const int blk = blockIdx.x;          // 0..511
    const int b   = blk >> 8;            // batch
    const int t   = blk & 255;
    const int i0  = (t >> 4) * 32;
    const int j0  = (t & 15) * 32;
    const int tid = threadIdx.x;

    for (int s = tid; s < 32 * HID_; s += 256) {
        const int r = s >> 7, cc = s & 127;
        se1[r][cc] = e12[(size_t)(b * N_ + i0 + r) * 256 + cc];
        se2[r][cc] = e12[(size_t)(b * N_ + j0 + r) * 256 + HID_ + cc];
    }
    if (tid < HID_) { sb1[tid] = b1[tid]; sw2[tid] = W2[tid]; }
    __syncthreads();

    const float bias2 = b2[0];
    const int li  = tid >> 3;            // 0..31
    const int lj0 = (tid & 7) * 4;       // 0,4,...,28

    float acc[4] = {0.f, 0.f, 0.f, 0.f};
    for (int hh = 0; hh < HID_; ++hh) {
        const float a1 = se1[li][hh] + sb1[hh];
        const float w  = sw2[hh];
#pragma unroll
        for (int q = 0; q < 4; ++q) {
            const float s = a1 + se2[lj0 + q][hh];
            acc[q] = fmaf(fmaxf(s, 0.f), w, acc[q]);
        }
    }

    float lsum = 0.f;
#pragma unroll
    for (int q = 0; q < 4; ++q) {
        float e = acc[q] + bias2;
        e = (e >= 0.f) ? e : 0.01f * e;                    // leaky relu
        const size_t gi = (size_t)b * N_ * N_ + (size_t)(i0 + li) * N_ + (j0 + lj0 + q);
        const float ev = adj[gi] ? __expf(e) : 0.f;        // mask -> exp(-1e30) = 0
        expE[gi] = ev;
        lsum += ev;
    }

    red[tid] = lsum;
    __syncthreads();
    for (int s = 128; s > 0; s >>= 1) {
        if (tid < s) red[tid] += red[tid + s];
        __syncthreads();
    }
    if (tid == 0) partials[blk] = red[0];
}

// ---------------------------------------------------------------------------
// Kernel 4: sumexp[b] = sum of 256 block partials (deterministic tree).
// ---------------------------------------------------------------------------
__global__ void k_reduce(const float* __restrict__ partials, float* __restrict__ sumexp) {
    __shared__ float red[256];
    red[threadIdx.x] = partials[blockIdx.x * 256 + threadIdx.x];
    __syncthreads();
    for (int s = 128; s > 0; s >>= 1) {
        if (threadIdx.x < s) red[threadIdx.x] += red[threadIdx.x + s];
        __syncthreads();
    }
    if (threadIdx.x == 0) sumexp[blockIdx.x] = red[0];
}

// ---------------------------------------------------------------------------
// Kernel 5: out[b,i,a] = (1/sumexp[b]) * sum_j expE[b,i,j] * h[b*N+j, a]
//   Per batch: M=512, K=512, Ncol=64.  f32 WMMA with scale folded in epilogue.
// ---------------------------------------------------------------------------
__global__ void k_out(const float* __restrict__ expE, const float* __restrict__ h,
                      const float* __restrict__ sumexp, float* __restrict__ out) {
    const int lane = threadIdx.x & 31;
    const int wave = threadIdx.x >> 5;
    const int tile = blockIdx.x * 4 + wave;   // 256 tiles = 2 batches x 32 x 4
    const int b  = tile >> 7;
    const int tm = (tile >> 2) & 31;
    const int tn = tile & 3;
    const int kb = (lane >> 4) * 2;
    const int m  = tm * 16 + (lane & 15);
    const int n  = tn * 16 + (lane & 15);

    const float* arow = expE + (size_t)b * N_ * N_ + (size_t)m * N_;
    const float* bcol = h    + (size_t)b * N_ * ATT_ + n;   // B[k][n] = h[b*N+k][n]

    v8f c = {};
    for (int k = 0; k < N_; k += 4) {
        v2f a, bb;
        a.x  = arow[k + kb];
        a.y  = arow[k + kb + 1];
        bb.x = bcol[(size_t)(k + kb) * ATT_];
        bb.y = bcol[(size_t)(k + kb + 1) * ATT_];
        c = __builtin_amdgcn_wmma_f32_16x16x4_f32(false, a, false, bb,
                                                  (short)0, c, false, false);
    }
    const float inv = 1.0f / sumexp[b];
    const int row0 = tm * 16 + (lane >> 4) * 8;
    const int col  = tn * 16 + (lane & 15);
#pragma unroll
    for (int v = 0; v < 8; ++v)
        out[((size_t)b * N_ + row0 + v) * ATT_ + col] = c[v] * inv;
}

// ---------------------------------------------------------------------------
extern "C" void kernel_launch(void* const* d_in, const int* in_sizes, int n_in,
                              void* d_out, int out_size, void* d_ws, size_t ws_size,
                              hipStream_t stream) {
    const float* x   = (const float*)d_in[0];
    const int*   adj = (const int*)  d_in[1];
    const float* Ww  = (const float*)d_in[2];
    const float* Wb  = (const float*)d_in[3];
    const float* W1  = (const float*)d_in[4];
    const float* b1  = (const float*)d_in[5];
    const float* W2  = (const float*)d_in[6];
    const float* b2  = (const float*)d_in[7];
    float* out = (float*)d_out;

    // Workspace layout (floats): ~3.3 MB total, lives in L2.
    float* ws    = (float*)d_ws;
    float* h_    = ws;                                // 1024 * 64
    float* e12   = h_   + (size_t)B_ * N_ * ATT_;     // 1024 * 256
    float* expE  = e12  + (size_t)B_ * N_ * 256;      // 2 * 512 * 512
    float* parts = expE + (size_t)B_ * N_ * N_;       // 512
    float* sums  = parts + 512;                       // 2

    k_h     <<< 64, 128, 0, stream>>>(x, Ww, Wb, h_);
    k_e12   <<<256, 128, 0, stream>>>(h_, W1, e12);
    k_pair  <<<512, 256, 0, stream>>>(e12, b1, W2, b2, adj, expE, parts);
    k_reduce<<<  2, 256, 0, stream>>>(parts, sums);
    k_out   <<< 64, 128, 0, stream>>>(expE, h_, sums, out);
}